// GATLayerTemporal_91242285236246
// MI455X (gfx1250) — compile-verified
//
#include <hip/hip_runtime.h>
#include <hip/hip_bf16.h>
#include <math.h>

typedef __attribute__((ext_vector_type(2))) float v2f;
typedef __attribute__((ext_vector_type(8))) float v8f;

#define N_NODES 100
#define HW      256   // 16*16 spatial
#define F_IN    8
#define F_OUT   16
#define ALPHA   0.2f

// ---------------------------------------------------------------------------
// Kernel A: Wh = x @ W   -> (25600 x 8) * (8 x 16), f32 WMMA 16x16x4, K=8
// Wh stored row-major (25600, 16): Wh[(m*256+p)*16 + f]
// ---------------------------------------------------------------------------
__global__ void wh_kernel(const float* __restrict__ x,
                          const float* __restrict__ W,
                          float* __restrict__ Wh) {
  const int lane = threadIdx.x;      // 0..31, one wave per 16-row tile
  const int half = lane >> 4;        // 0 or 1
  const int lm   = lane & 15;
  const int row  = blockIdx.x * 16 + lm;   // < 25600 exactly (1600 tiles)

  v8f acc = {};
#pragma unroll
  for (int kb = 0; kb < F_IN; kb += 4) {
    const int k0 = kb + half * 2;
    v2f a, b;
    a.x = x[row * F_IN + k0];
    a.y = x[row * F_IN + k0 + 1];
    b.x = W[(k0)     * F_OUT + lm];
    b.y = W[(k0 + 1) * F_OUT + lm];
    acc = __builtin_amdgcn_wmma_f32_16x16x4_f32(false, a, false, b,
                                                (short)0, acc, false, false);
  }
#pragma unroll
  for (int r = 0; r < 8; ++r) {
    const int orow = blockIdx.x * 16 + r + half * 8;
    Wh[orow * F_OUT + lm] = acc[r];
  }
}

// ---------------------------------------------------------------------------
// Kernel B: d[i,y,u] = dot(a[u*256 : u*256+256], Wh[i*4096 + y*256 : +256])
// 3200 contiguous 256-length dots; replaces the 328MB a_input tensor.
// ---------------------------------------------------------------------------
__global__ void d_kernel(const float* __restrict__ avec,
                         const float* __restrict__ Wh,
                         float* __restrict__ dvec) {
  const int idx = blockIdx.x * blockDim.x + threadIdx.x;
  if (idx >= N_NODES * 32) return;
  const int i = idx >> 5;
  const int r = idx & 31;
  const int y = r >> 1;
  const int u = r & 1;
  const float* ap = avec + u * 256;
  const float* wp = Wh + i * 4096 + y * 256;
  float s = 0.f;
#pragma unroll 8
  for (int t = 0; t < 256; ++t) s = fmaf(ap[t], wp[t], s);
  dvec[idx] = s;
}

// ---------------------------------------------------------------------------
// Kernel C: e -> leakyrelu -> adj mask -> softmax over m -> att[i,m,k]
// One wave per (i,k); each lane owns m = lane + 32*j, wave32 shuffle reduce.
// ---------------------------------------------------------------------------
__global__ void att_kernel(const float* __restrict__ dvec,
                           const int* __restrict__ adj,
                           float* __restrict__ att) {
  const int bid  = blockIdx.x;       // 0..1599
  const int i    = bid >> 4;
  const int k    = bid & 15;
  const int lane = threadIdx.x;      // 0..31

  float ev[4];
  float mx = -3.0e38f;
#pragma unroll
  for (int j = 0; j < 4; ++j) {
    const int m = lane + 32 * j;
    float e = -3.0e38f;
    if (m < N_NODES) {
      if (k < 8) {
        const int s0 = 16 * m + 2 * k;
        const int y0 = s0 / 100;
        const int y1 = (s0 + 1) / 100;
        e = dvec[i * 32 + y0 * 2 + 0] + dvec[i * 32 + y1 * 2 + 1];
      } else {
        const int kk = k - 8;
        e = dvec[m * 32 + (2 * kk) * 2 + 0] + dvec[m * 32 + (2 * kk + 1) * 2 + 1];
      }
      e = (e > 0.f) ? e : ALPHA * e;                    // LeakyReLU(0.2)
      e = (adj[i * N_NODES + m] > 0) ? e : -9.0e15f;    // mask
      mx = fmaxf(mx, e);
    }
    ev[j] = e;
  }
  for (int off = 16; off > 0; off >>= 1) mx = fmaxf(mx, __shfl_xor(mx, off, 32));

  float ex[4];
  float s = 0.f;
#pragma unroll
  for (int j = 0; j < 4; ++j) {
    const int m = lane + 32 * j;
    if (m < N_NODES) { ex[j] = expf(ev[j] - mx); s += ex[j]; }
  }
  for (int off = 16; off > 0; off >>= 1) s += __shfl_xor(s, off, 32);
  const float inv = 1.f / s;

#pragma unroll
  for (int j = 0; j < 4; ++j) {
    const int m = lane + 32 * j;
    if (m < N_NODES) att[(i * N_NODES + m) * F_OUT + k] = ex[j] * inv;
  }
}

// ---------------------------------------------------------------------------
// Kernel D: h'[i,p,f] = sum_m att[i,m,f] * Wh[m,p,f]
// 16 f-batched GEMMs (100x100)*(100x256) via f32 WMMA 16x16x4, K=100 (25 steps)
// Grid: (nt=16, mt=7, f=16); one wave per output tile.
// ---------------------------------------------------------------------------
__global__ void hprime_kernel(const float* __restrict__ att,
                              const float* __restrict__ Wh,
                              float* __restrict__ out) {
  const int nt   = blockIdx.x;       // p tile 0..15
  const int mt   = blockIdx.y;       // i tile 0..6 (rows padded 100 -> 112)
  const int f    = blockIdx.z;       // 0..15
  const int lane = threadIdx.x;
  const int half = lane >> 4;
  const int lm   = lane & 15;

  const int   i_row    = mt * 16 + lm;
  const float rowValid = (i_row < N_NODES) ? 1.0f : 0.0f;   // no divergence
  const int   i_c      = (i_row < N_NODES) ? i_row : (N_NODES - 1);
  const int   p        = nt * 16 + lm;

  v8f acc = {};
  for (int kb = 0; kb < N_NODES; kb += 4) {   // 100 = 25*4 exactly
    const int m0 = kb + half * 2;
    v2f a, b;
    a.x = att[(i_c * N_NODES + m0)     * F_OUT + f] * rowValid;
    a.y = att[(i_c * N_NODES + m0 + 1) * F_OUT + f] * rowValid;
    b.x = Wh[(m0 * HW + p)       * F_OUT + f];
    b.y = Wh[((m0 + 1) * HW + p) * F_OUT + f];
    acc = __builtin_amdgcn_wmma_f32_16x16x4_f32(false, a, false, b,
                                                (short)0, acc, false, false);
  }
#pragma unroll
  for (int r = 0; r < 8; ++r) {
    const int orow = mt * 16 + r + half * 8;
    if (orow < N_NODES)
      out[(orow * HW + p) * F_OUT + f] = acc[r];
  }
}

// ---------------------------------------------------------------------------
extern "C" void kernel_launch(void* const* d_in, const int* in_sizes, int n_in,
                              void* d_out, int out_size, void* d_ws, size_t ws_size,
                              hipStream_t stream) {
  (void)in_sizes; (void)n_in; (void)out_size; (void)ws_size;

  const float* x    = (const float*)d_in[0];   // (100,16,16,8)
  const int*   adj  = (const int*)  d_in[1];   // (100,100)
  const float* W    = (const float*)d_in[2];   // (8,16)
  const float* avec = (const float*)d_in[3];   // (512,1)
  float* out = (float*)d_out;                  // (100,16,16,16)

  float* ws   = (float*)d_ws;
  float* Wh   = ws;                        // 409600 floats (1.6 MB)
  float* attp = ws + 409600;               // 160000 floats (0.64 MB)
  float* dv   = ws + 409600 + 160000;      // 3200 floats

  wh_kernel    <<<1600, 32, 0, stream>>>(x, W, Wh);
  d_kernel     <<<25, 128, 0, stream>>>(avec, Wh, dv);
  att_kernel   <<<1600, 32, 0, stream>>>(dv, adj, attp);
  hprime_kernel<<<dim3(16, 7, 16), 32, 0, stream>>>(attp, Wh, out);
}